// BaseLossEvaluator_76682346103502
// MI455X (gfx1250) — compile-verified
//
#include <hip/hip_runtime.h>

typedef float v8f __attribute__((ext_vector_type(8)));
typedef float v2f __attribute__((ext_vector_type(2)));
typedef __attribute__((address_space(3))) float lds_f;

#define GAMMA_C  0.99f
#define LAMBDA_C 0.95f
#define GL_C     (GAMMA_C * LAMBDA_C)
#define T_LEN    1024
#define BATCH_N  32768
#define CHUNK    16
#define NCHUNK   64            // T_LEN / CHUNK
#define COLS     128           // columns per block (8 waves * 16)
#define LSTRIDE  136           // padded LDS row stride (floats): 2*136 mod 64 = 16
#define VROWS    17            // values rows per chunk (need V[s+1])
#define VOFF     (VROWS * LSTRIDE)
#define BUF_FLOATS (VOFF + CHUNK * LSTRIDE)   // 4488 floats per buffer

// Per-lane 16B async copy global -> LDS (ASYNCcnt path, bypasses VGPRs).
// VDST operand is a real addrspace(3) pointer (= 32-bit LDS byte offset), so the
// shared array escapes and the compiler must honor the asm's LDS writes.
__device__ __forceinline__ void async_copy_b128(lds_f* lp, const float* gp) {
    asm volatile("global_load_async_to_lds_b128 %0, %1, off"
                 :: "v"(lp), "v"(gp)
                 : "memory");
}

__device__ __forceinline__ void wait_async_zero() {
#if __has_builtin(__builtin_amdgcn_s_wait_asynccnt)
    __builtin_amdgcn_s_wait_asynccnt(0);
#else
    asm volatile("s_wait_asynccnt 0x0" ::: "memory");
#endif
}

// Stage one 16-step chunk (values 17x128 at buf, rewards 16x128 at buf+VOFF)
// via per-lane b128 async copies (32 lanes x 16B = 512B = one full row segment).
__device__ __forceinline__ void stage_chunk(int c, lds_f* buf,
                                            const float* __restrict__ rewards,
                                            const float* __restrict__ values,
                                            int colbase, int tid) {
    const int rbase = c * CHUNK;
    for (int idx = tid; idx < VROWS * 32; idx += 256) {
        int r = idx >> 5, g = (idx & 31) << 2;       // g: column offset in floats
        int gr = rbase + r;
        if (gr < T_LEN)
            async_copy_b128(buf + (r * LSTRIDE + g),
                            &values[(size_t)gr * BATCH_N + colbase + g]);
    }
    for (int idx = tid; idx < CHUNK * 32; idx += 256) {
        int r = idx >> 5, g = (idx & 31) << 2;
        async_copy_b128(buf + (VOFF + r * LSTRIDE + g),
                        &rewards[(size_t)(rbase + r) * BATCH_N + colbase + g]);
    }
}

__global__ __launch_bounds__(256)
void gae_wmma_async_kernel(const float* __restrict__ rewards,
                           const float* __restrict__ values,
                           float* __restrict__ out) {
    __shared__ float lds_raw[2 * BUF_FLOATS];        // double-buffered tiles
    lds_f* lds3 = (lds_f*)lds_raw;                   // addrspace(3) view

    const int tid  = threadIdx.x;
    const int lane = tid & 31;
    const int half = lane >> 4;                      // 0: lanes 0-15, 1: lanes 16-31
    const int coln = lane & 15;                      // N within 16-wide tile
    const int wave = tid >> 5;                       // 0..7
    const int colbase = blockIdx.x * COLS;
    const int wcol = wave * CHUNK + coln;            // lane's column within block
    const int gcol = colbase + wcol;

    // A matrix (chunk-invariant): A[i][k] = gl^(k-i) for k>=i else 0.
    // f32 16x4 A layout: M = lane%16; VGPR v holds K = 2*half + v per 4-wide slice.
    v2f Amat[4];
#pragma unroll
    for (int kb = 0; kb < 4; ++kb) {
#pragma unroll
        for (int v = 0; v < 2; ++v) {
            int K = 4 * kb + 2 * half + v;
            int d = K - coln;
            float pw = 0.0f;
            if (d >= 0) { pw = 1.0f; for (int i = 0; i < d; ++i) pw *= GL_C; }
            Amat[kb][v] = pw;
        }
    }
    // Carry scale: acc register j holds row (j + 8*half); factor gl^(16-row).
    float cpow[8];
#pragma unroll
    for (int j = 0; j < 8; ++j) {
        int e = 16 - (j + 8 * half);
        float pw = 1.0f;
        for (int i = 0; i < e; ++i) pw *= GL_C;
        cpow[j] = pw;
    }

    // Prologue: fill buffer 0 with the latest-in-time chunk.
    stage_chunk(NCHUNK - 1, lds3, rewards, values, colbase, tid);
    wait_async_zero();
    __syncthreads();

    float carry = 0.0f;
    unsigned p = 0;

    for (int c = NCHUNK - 1; c >= 0; --c) {
        const int rbase = c * CHUNK;
        lds_f* curbuf = lds3 + p * BUF_FLOATS;
        lds_f* nxtbuf = lds3 + (p ^ 1u) * BUF_FLOATS;

        // Kick off async fill of the other buffer while we compute this one.
        if (c > 0) stage_chunk(c - 1, nxtbuf, rewards, values, colbase, tid);

        lds_f* vbuf = curbuf;                // values tile
        lds_f* dbuf = curbuf + VOFF;         // rewards tile (deltas on the fly)

        // Accumulator seeded with carry term: C[row][col] = gl^(16-row)*carry[col].
        v8f acc;
#pragma unroll
        for (int j = 0; j < 8; ++j) acc[j] = cpow[j] * carry;

        // Build B with delta formed on the fly; 4 chained f32 WMMAs (K=16).
#pragma unroll
        for (int kb = 0; kb < 4; ++kb) {
            v2f Bv;
#pragma unroll
            for (int v = 0; v < 2; ++v) {
                int K = 4 * kb + 2 * half + v;       // B layout: K = 2*half + v per slice
                float dv = dbuf[K * LSTRIDE + wcol]
                         + GAMMA_C * vbuf[(K + 1) * LSTRIDE + wcol]
                         - vbuf[K * LSTRIDE + wcol];
                Bv[v] = (rbase + K < T_LEN - 1) ? dv : 0.0f;   // pad delta[1023]=0
            }
            acc = __builtin_amdgcn_wmma_f32_16x16x4_f32(
                false, Amat[kb], false, Bv, (short)0, acc, false, false);
        }

        // Store D (VGPR j -> row j + 8*half); mask padded t = 1023; NT streaming stores.
#pragma unroll
        for (int j = 0; j < 8; ++j) {
            int t = rbase + j + 8 * half;
            if (t < T_LEN - 1)
                __builtin_nontemporal_store(acc[j], &out[(size_t)t * BATCH_N + gcol]);
        }

        // Carry = adv at this chunk's first row = D row 0 (acc[0] of lanes 0-15).
        carry = __shfl(acc[0], coln, 32);

        // Next buffer complete + all waves done reading current buffer.
        wait_async_zero();
        __syncthreads();
        p ^= 1u;
    }
}

extern "C" void kernel_launch(void* const* d_in, const int* in_sizes, int n_in,
                              void* d_out, int out_size, void* d_ws, size_t ws_size,
                              hipStream_t stream) {
    const float* rewards = (const float*)d_in[0];
    const float* values  = (const float*)d_in[1];
    float* out = (float*)d_out;
    dim3 grid(BATCH_N / COLS);   // 256 blocks
    dim3 block(256);             // 8 wave32 per block
    hipLaunchKernelGGL(gae_wmma_async_kernel, grid, block, 0, stream, rewards, values, out);
}